// GwcVolume_85048942395514
// MI455X (gfx1250) — compile-verified
//
#include <hip/hip_runtime.h>

typedef float v2f __attribute__((ext_vector_type(2)));
typedef float v8f __attribute__((ext_vector_type(8)));

#define BATCH   2
#define CHAN    320
#define GROUPS  40
#define CPG     8
#define HDIM    96
#define WDIM    192
#define MAXD    48
#define HW      (HDIM * WDIM)
#define WTILES  (WDIM / 16)   // 12

// One wave32 per (b, g, h, w_tile). Computes the 16x64 band slice of
// M = (L/8)^T R (K = cpg = 8) via fp32 WMMA and scatters it into the
// cost volume with constant-stride (HW+1) stores off one base pointer.
__global__ __launch_bounds__(256)
void gwc_volume_wmma(const float* __restrict__ L,
                     const float* __restrict__ R,
                     float* __restrict__ out)
{
    const int lane  = threadIdx.x & 31;
    const int wid   = blockIdx.x * 8 + (threadIdx.x >> 5);

    int t = wid;
    const int wt = t % WTILES;  t /= WTILES;
    const int h  = t % HDIM;    t /= HDIM;
    const int g  = t % GROUPS;  t /= GROUPS;
    const int b  = t;

    const int col   = lane & 15;     // matrix column within tile (w for A, j for B)
    const int khalf = lane >> 4;     // half-wave selects upper K pair
    const int w_base = wt * 16;

    const float* lb = L + ((size_t)b * CHAN + (size_t)g * CPG) * HW + (size_t)h * WDIM;
    const float* rb = R + ((size_t)b * CHAN + (size_t)g * CPG) * HW + (size_t)h * WDIM;
    float*       ob = out + ((size_t)b * GROUPS + g) * (size_t)MAXD * HW + (size_t)h * WDIM;

    // A operands: left tile, 16 w-columns x 8 channels -> two K=4 steps.
    // Layout (32-bit A 16x4): vgpr v, lane l holds A[M = l%16, K = v + 2*(l/16)].
    // Fold the group-mean 1/8 into A so the epilogue stores raw accumulators.
    v2f a0, a1;
    {
        const float* p = lb + w_base + col;
        a0.x = p[(size_t)(2 * khalf + 0) * HW] * 0.125f;
        a0.y = p[(size_t)(2 * khalf + 1) * HW] * 0.125f;
        a1.x = p[(size_t)(4 + 2 * khalf + 0) * HW] * 0.125f;
        a1.y = p[(size_t)(4 + 2 * khalf + 1) * HW] * 0.125f;
    }

    const int w0 = w_base + 8 * khalf;   // w-row of acc element 0 for this lane

    // 4 j-tiles cover j in [w_base-48, w_base+15] => d = w - j in [0,48) fully covered.
    #pragma unroll
    for (int jt = 0; jt < 4; ++jt) {
        const int j_base = w_base - 48 + 16 * jt;
        const int j      = j_base + col;        // may be negative at the left edge
        const int jc     = (j < 0) ? 0 : j;     // clamped load address

        // B operands: right tile, 4x16 K-major; same lane mapping as A.
        v2f b0, b1;
        {
            const float* p = rb + jc;
            b0.x = p[(size_t)(2 * khalf + 0) * HW];
            b0.y = p[(size_t)(2 * khalf + 1) * HW];
            b1.x = p[(size_t)(4 + 2 * khalf + 0) * HW];
            b1.y = p[(size_t)(4 + 2 * khalf + 1) * HW];
        }

        // Wave-uniform: only left-edge tiles have j<0 lanes. Zero B there so the
        // accumulator is exactly 0 for the reference's masked (w<d) region.
        if (j_base < 0) {
            const float m = (j >= 0) ? 1.0f : 0.0f;
            b0.x *= m; b0.y *= m; b1.x *= m; b1.y *= m;
        }

        v8f acc = {};
        acc = __builtin_amdgcn_wmma_f32_16x16x4_f32(
                  false, a0, false, b0, (short)0, acc, false, false);
        acc = __builtin_amdgcn_wmma_f32_16x16x4_f32(
                  false, a1, false, b1, (short)0, acc, false, false);

        // D layout: vgpr r, lane holds (M = r + 8*khalf, N = col) = (w-row, j-col).
        // d = w - j = d0 + r; address advances by (HW+1) elements per r, so all 8
        // stores share one base pointer with immediate offsets.
        const int d0 = w0 - j;
        float* p = ob + (ptrdiff_t)d0 * HW + w0;
        #pragma unroll
        for (int r = 0; r < 8; ++r) {
            if ((unsigned)(d0 + r) < (unsigned)MAXD) {
                p[(ptrdiff_t)r * (HW + 1)] = acc[r];
            }
        }
    }
}

extern "C" void kernel_launch(void* const* d_in, const int* in_sizes, int n_in,
                              void* d_out, int out_size, void* d_ws, size_t ws_size,
                              hipStream_t stream) {
    const float* Lf = (const float*)d_in[0];
    const float* Rf = (const float*)d_in[1];
    float* out = (float*)d_out;

    const int total_waves = BATCH * GROUPS * HDIM * WTILES;  // 92160
    const int blocks = total_waves / 8;                      // 11520 (exact)
    gwc_volume_wmma<<<dim3(blocks), dim3(256), 0, stream>>>(Lf, Rf, out);
    (void)in_sizes; (void)n_in; (void)out_size; (void)d_ws; (void)ws_size;
}